// MambaInjectionBlock_49615462203507
// MI455X (gfx1250) — compile-verified
//
#include <hip/hip_runtime.h>
#include <math.h>

typedef __attribute__((ext_vector_type(2))) float v2f;
typedef __attribute__((ext_vector_type(8))) float v8f;

#define L_TOK 4096
#define CDIM  64

// ---------------------------------------------------------------------------
// fp32 WMMA GEMM: Cout[M x N] = epilogue(A[M x K] @ B^T (+bias)) (+resid)
// B is ALWAYS (N x K) row-major (weights are pre-transposed on device), so
// every fragment is one contiguous 8-byte load with immediate offsets.
// Each wave computes a 16x32 tile: one shared A fragment feeds two
// independent v_wmma_f32_16x16x4_f32 accumulator chains (hides acc RAW).
// Distance-1 software pipeline overlaps loads with the matrix pipe.
// act: 0 = none, 1 = exact GELU.
// ---------------------------------------------------------------------------
__global__ __launch_bounds__(128)
void gemm_wmma_f32(const float* __restrict__ A, const float* __restrict__ B,
                   const float* __restrict__ bias, const float* __restrict__ resid,
                   float* __restrict__ Cout, int M, int N, int K, int act) {
  int lane = threadIdx.x & 31;
  int wv   = threadIdx.x >> 5;
  int half = lane >> 4;            // K-half per ISA fragment layout
  int lm   = lane & 15;
  int n0 = blockIdx.x * 32;
  int m0 = (blockIdx.y * 4 + wv) * 16;
  if (m0 >= M) return;             // uniform per wave -> EXEC stays full
  int  nnA = n0 + lm,  nnB = n0 + 16 + lm;
  bool inA = nnA < N,  inB = nnB < N;
  float mA = inA ? 1.f : 0.f,  mB = inB ? 1.f : 0.f;  // mask-mul (clamped
  int   ncA = inA ? nnA : 0,   ncB = inB ? nnB : 0;   // addrs stay valid)

  v8f acc0 = {}, acc1 = {};
  const float* Ap = A + (size_t)(m0 + lm) * K + 2 * half;
  const float* B0 = B + (size_t)ncA * K + 2 * half;
  const float* B1 = B + (size_t)ncB * K + 2 * half;

  v2f a  = *(const v2f*)Ap;
  v2f b0 = *(const v2f*)B0;
  v2f b1 = *(const v2f*)B1;
#pragma unroll 4
  for (int k0 = 4; k0 < K; k0 += 4) {
    v2f an  = *(const v2f*)(Ap + k0);     // prefetch next fragments; the
    v2f b0n = *(const v2f*)(B0 + k0);     // wait lands after the WMMAs
    v2f b1n = *(const v2f*)(B1 + k0);
    v2f c0, c1;
    c0.x = b0.x * mA;  c0.y = b0.y * mA;
    c1.x = b1.x * mB;  c1.y = b1.y * mB;
    acc0 = __builtin_amdgcn_wmma_f32_16x16x4_f32(false, a, false, c0, (short)0,
                                                 acc0, false, false);
    acc1 = __builtin_amdgcn_wmma_f32_16x16x4_f32(false, a, false, c1, (short)0,
                                                 acc1, false, false);
    a = an;  b0 = b0n;  b1 = b1n;
  }
  {
    v2f c0, c1;
    c0.x = b0.x * mA;  c0.y = b0.y * mA;
    c1.x = b1.x * mB;  c1.y = b1.y * mB;
    acc0 = __builtin_amdgcn_wmma_f32_16x16x4_f32(false, a, false, c0, (short)0,
                                                 acc0, false, false);
    acc1 = __builtin_amdgcn_wmma_f32_16x16x4_f32(false, a, false, c1, (short)0,
                                                 acc1, false, false);
  }
  auto store_half = [&](v8f& acc, int nn, bool in) {
    if (!in) return;
    float bs = bias ? bias[nn] : 0.f;
#pragma unroll
    for (int r = 0; r < 8; ++r) {
      int m = m0 + r + 8 * half;   // C/D layout: lanes16-31 hold M+8
      float v = acc[r] + bs;
      if (act == 1) v = 0.5f * v * (1.f + erff(v * 0.70710678118654752f));
      if (resid) v += resid[(size_t)m * N + nn];
      Cout[(size_t)m * N + nn] = v;
    }
  };
  store_half(acc0, nnA, inA);
  store_half(acc1, nnB, inB);
}

// Wt[n*K + k] = W[k*N + n]  (tiny one-time weight transposes)
__global__ void transpose_kernel(const float* __restrict__ W, float* __restrict__ Wt,
                                 int K, int N) {
  int i = blockIdx.x * blockDim.x + threadIdx.x;
  if (i >= K * N) return;
  int n = i % N, k = i / N;
  Wt[(size_t)n * K + k] = W[i];
}

// ---------------------------------------------------------------------------
// LayerNorm over Cd per token (one wave32 per token), optional silu(gate) mul.
// ---------------------------------------------------------------------------
__global__ __launch_bounds__(128)
void ln_gate_kernel(const float* __restrict__ in, const float* __restrict__ g,
                    const float* __restrict__ b, const float* __restrict__ gate,
                    int gstride, int goff, float eps,
                    float* __restrict__ out, int L, int Cd) {
  int lane = threadIdx.x & 31;
  int wv   = threadIdx.x >> 5;
  int t = blockIdx.x * 4 + wv;
  if (t >= L) return;
  const float* row = in + (size_t)t * Cd;
  float s = 0.f, ss = 0.f;
  for (int c = lane; c < Cd; c += 32) { float v = row[c]; s += v; ss += v * v; }
  for (int off = 16; off; off >>= 1) {
    s  += __shfl_xor(s,  off, 32);
    ss += __shfl_xor(ss, off, 32);
  }
  float mean = s / (float)Cd;
  float var  = ss / (float)Cd - mean * mean;
  float inv  = rsqrtf(var + eps);
  for (int c = lane; c < Cd; c += 32) {
    float v = (row[c] - mean) * inv * g[c] + b[c];
    if (gate) {
      float z = gate[(size_t)t * gstride + goff + c];
      v *= z / (1.f + expf(-z));       // * silu(z)
    }
    out[(size_t)t * Cd + c] = v;
  }
}

// ---------------------------------------------------------------------------
// Depthwise 7x7 SAME conv + SiLU on first di channels of xz (L x 2di).
// windowed: pad within 8x8 windows; else within full 64x64 image.
// ---------------------------------------------------------------------------
__global__ void dwconv7_silu(const float* __restrict__ xz, const float* __restrict__ w,
                             const float* __restrict__ b, float* __restrict__ out,
                             int di, int windowed) {
  int i = blockIdx.x * blockDim.x + threadIdx.x;
  if (i >= L_TOK * di) return;
  int d = i % di, t = i / di;
  int h = t >> 6, wq = t & 63;
  const float* wt = w + (size_t)d * 49;
  float acc = b[d];
  int st = 2 * di;
  if (!windowed) {
    for (int ky = 0; ky < 7; ++ky) {
      int hh = h + ky - 3; if (hh < 0 || hh >= 64) continue;
      for (int kx = 0; kx < 7; ++kx) {
        int ww = wq + kx - 3; if (ww < 0 || ww >= 64) continue;
        acc += xz[(size_t)(hh * 64 + ww) * st + d] * wt[ky * 7 + kx];
      }
    }
  } else {
    int r = h & 7, c = wq & 7, hb = h - r, wb = wq - c;
    for (int ky = 0; ky < 7; ++ky) {
      int rr = r + ky - 3; if (rr < 0 || rr >= 8) continue;
      for (int kx = 0; kx < 7; ++kx) {
        int cc = c + kx - 3; if (cc < 0 || cc >= 8) continue;
        acc += xz[(size_t)((hb + rr) * 64 + wb + cc) * st + d] * wt[ky * 7 + kx];
      }
    }
  }
  out[i] = acc / (1.f + expf(-acc));   // silu
}

// ---------------------------------------------------------------------------
// Selective scan: one thread per (window, dir, channel). Fuses dt-proj +
// softplus; direction reversal/transpose folded into token index so outputs
// land at the true token position (no un-permute pass needed).
// dbl rows: [4 dirs][ dts(R) | Bs(NST) | Cs(NST) ]
// ---------------------------------------------------------------------------
template <int NST>
__global__ void scan_kernel(const float* __restrict__ dbl, const float* __restrict__ xxc,
                            const float* __restrict__ dtw, const float* __restrict__ dtb,
                            const float* __restrict__ Alog, const float* __restrict__ Dp,
                            float* __restrict__ ybuf, int di, int windowed) {
  const int R = 4;
  int tid = blockIdx.x * blockDim.x + threadIdx.x;
  int d = tid % di;
  int rest = tid / di;
  int nk = windowed ? 256 : 4;        // (#windows * 4 dirs) or 4 dirs
  if (rest >= nk) return;
  int k = rest & 3;
  int win = rest >> 2;
  int NT = 4 * (R + 2 * NST);
  int kb = k * (R + 2 * NST);
  int pd = k * di + d;
  float w0 = dtw[pd * 4 + 0], w1 = dtw[pd * 4 + 1];
  float w2 = dtw[pd * 4 + 2], w3 = dtw[pd * 4 + 3];
  float bdt = dtb[pd];
  float Dv  = Dp[pd];
  float Aa[NST];
#pragma unroll
  for (int n = 0; n < NST; ++n) Aa[n] = -expf(Alog[(size_t)pd * NST + n]);
  float hst[NST];
#pragma unroll
  for (int n = 0; n < NST; ++n) hst[n] = 0.f;
  int Lsc = windowed ? 64 : L_TOK;
  int wy = win >> 3, wx = win & 7;
  for (int l = 0; l < Lsc; ++l) {
    int t;
    if (windowed) {
      int li = (k >= 2) ? (63 - l) : l;
      int r_, c_;
      if (k & 1) { c_ = li >> 3; r_ = li & 7; } else { r_ = li >> 3; c_ = li & 7; }
      t = (wy * 8 + r_) * 64 + (wx * 8 + c_);
    } else {
      int li = (k >= 2) ? (L_TOK - 1 - l) : l;
      int hh, ww;
      if (k & 1) { ww = li >> 6; hh = li & 63; } else { hh = li >> 6; ww = li & 63; }
      t = hh * 64 + ww;
    }
    const float* pr = dbl + (size_t)t * NT + kb;
    float dtraw = pr[0] * w0 + pr[1] * w1 + pr[2] * w2 + pr[3] * w3 + bdt;
    float dt = dtraw > 20.f ? dtraw : log1pf(expf(dtraw));  // softplus
    float xv = xxc[(size_t)t * di + d];
    float dtx = dt * xv;
    float y = Dv * xv;
#pragma unroll
    for (int n = 0; n < NST; ++n) {
      float a = expf(dt * Aa[n]);
      hst[n] = hst[n] * a + dtx * pr[R + n];
      y += hst[n] * pr[R + NST + n];
    }
    ybuf[((size_t)k * L_TOK + t) * di + d] = y;
  }
}

__global__ void combine4(const float* __restrict__ yb, float* __restrict__ ym, int di) {
  int i = blockIdx.x * blockDim.x + threadIdx.x;
  size_t stride = (size_t)L_TOK * di;
  if (i >= (int)stride) return;
  ym[i] = yb[i] + yb[i + stride] + yb[i + 2 * stride] + yb[i + 3 * stride];
}

// ------------------------- prep path kernels -------------------------------
__global__ void bilerp_kernel(const float* __restrict__ cond, float* __restrict__ ci) {
  int i = blockIdx.x * blockDim.x + threadIdx.x;
  if (i >= L_TOK * 32) return;
  int c = i & 31, t = i >> 5;
  int h = t >> 6, w = t & 63;
  float fy = h * (31.f / 63.f), fx = w * (31.f / 63.f);
  int y0 = (int)fy; float ay = fy - y0; int y1 = y0 + 1 < 31 ? y0 + 1 : 31;
  int x0 = (int)fx; float ax = fx - x0; int x1 = x0 + 1 < 31 ? x0 + 1 : 31;
  const float* pc = cond + (size_t)c * 1024;
  float r0 = pc[y0 * 32 + x0] * (1.f - ay) + pc[y1 * 32 + x0] * ay;
  float r1 = pc[y0 * 32 + x1] * (1.f - ay) + pc[y1 * 32 + x1] * ay;
  ci[(size_t)t * 32 + c] = r0 * (1.f - ax) + r1 * ax;
}

__global__ void conv1x1_kernel(const float* __restrict__ ci, const float* __restrict__ w1,
                               const float* __restrict__ b1, float* __restrict__ t1) {
  int i = blockIdx.x * blockDim.x + threadIdx.x;
  if (i >= L_TOK * CDIM) return;
  int cc = i & 63, t = i >> 6;
  const float* wr = w1 + (size_t)cc * 32;
  const float* cr = ci + (size_t)t * 32;
  float acc = b1[cc];
  for (int k = 0; k < 32; ++k) acc += cr[k] * wr[k];
  t1[i] = acc;
}

__global__ void dw3_add_kernel(const float* __restrict__ t1, const float* __restrict__ w2,
                               const float* __restrict__ b2, const float* __restrict__ feat,
                               float* __restrict__ x0) {
  int i = blockIdx.x * blockDim.x + threadIdx.x;
  if (i >= L_TOK * CDIM) return;
  int cc = i & 63, t = i >> 6;
  int h = t >> 6, w = t & 63;
  const float* wt = w2 + (size_t)cc * 9;
  float acc = b2[cc];
  for (int ky = 0; ky < 3; ++ky) {
    int hh = h + ky - 1; if (hh < 0 || hh >= 64) continue;
    for (int kx = 0; kx < 3; ++kx) {
      int ww = w + kx - 1; if (ww < 0 || ww >= 64) continue;
      acc += t1[(size_t)(hh * 64 + ww) * 64 + cc] * wt[ky * 3 + kx];
    }
  }
  x0[i] = feat[i] + acc;
}

__global__ void enh_combine_kernel(const float* __restrict__ xl, const float* __restrict__ enh,
                                   const float* __restrict__ x0, float* __restrict__ y) {
  int i = blockIdx.x * blockDim.x + threadIdx.x;
  if (i >= L_TOK * CDIM) return;
  y[i] = xl[i] * enh[i & 63] + x0[i];
}

// ------------------------- host orchestration ------------------------------
struct BlockP {
  const float *ln1_g, *ln1_b, *inproj, *convw, *convb, *xproj, *dtw, *dtb,
              *Alog, *D, *ong, *onb, *outw, *ln2_g, *ln2_b, *fc1w, *fc1b,
              *fc2w, *fc2b;
  // device-transposed (N x K) copies
  const float *inprojT, *outwT, *fc1T, *fc2T;
};

static void run_block(const float* Xin, float* Xout, const BlockP& p,
                      int di, int NST, int mrC, int windowed,
                      float* ln1, float* xz, float* xxc, float* dbl,
                      float* yb, float* ym, float* ya, float* x1,
                      float* h0, float* h1, hipStream_t stream) {
  const int L = L_TOK, C = CDIM, R = 4;
  int NT = 4 * (R + 2 * NST);
  // LN1
  ln_gate_kernel<<<L / 4, 128, 0, stream>>>(Xin, p.ln1_g, p.ln1_b, nullptr, 0, 0,
                                            1e-6f, ln1, L, C);
  // in_proj (WMMA)
  gemm_wmma_f32<<<dim3((2 * di + 31) / 32, L / 64), 128, 0, stream>>>(
      ln1, p.inprojT, nullptr, nullptr, xz, L, 2 * di, C, 0);
  // depthwise 7x7 + SiLU
  int n1 = L * di;
  dwconv7_silu<<<(n1 + 255) / 256, 256, 0, stream>>>(xz, p.convw, p.convb, xxc, di, windowed);
  // batched 4-direction x_proj (WMMA; xproj_w is already (NT x di))
  gemm_wmma_f32<<<dim3((NT + 31) / 32, L / 64), 128, 0, stream>>>(
      xxc, p.xproj, nullptr, nullptr, dbl, L, NT, di, 0);
  // selective scans (dt-proj + softplus fused)
  int nthr = (windowed ? 256 : 4) * di;
  if (NST == 16)
    scan_kernel<16><<<(nthr + 127) / 128, 128, 0, stream>>>(dbl, xxc, p.dtw, p.dtb,
                                                            p.Alog, p.D, yb, di, windowed);
  else
    scan_kernel<8><<<(nthr + 127) / 128, 128, 0, stream>>>(dbl, xxc, p.dtw, p.dtb,
                                                           p.Alog, p.D, yb, di, windowed);
  combine4<<<(n1 + 255) / 256, 256, 0, stream>>>(yb, ym, di);
  // out-norm * silu(z)
  ln_gate_kernel<<<L / 4, 128, 0, stream>>>(ym, p.ong, p.onb, xz, 2 * di, di,
                                            1e-5f, ya, L, di);
  // out_proj + residual (WMMA)
  gemm_wmma_f32<<<dim3((C + 31) / 32, L / 64), 128, 0, stream>>>(
      ya, p.outwT, nullptr, Xin, x1, L, C, di, 0);
  // LN2
  ln_gate_kernel<<<L / 4, 128, 0, stream>>>(x1, p.ln2_g, p.ln2_b, nullptr, 0, 0,
                                            1e-6f, h0, L, C);
  // fc1 + GELU (WMMA)
  gemm_wmma_f32<<<dim3((mrC + 31) / 32, L / 64), 128, 0, stream>>>(
      h0, p.fc1T, p.fc1b, nullptr, h1, L, mrC, C, 1);
  // fc2 + residual (WMMA)
  gemm_wmma_f32<<<dim3((C + 31) / 32, L / 64), 128, 0, stream>>>(
      h1, p.fc2T, p.fc2b, x1, Xout, L, C, mrC, 0);
}

static void transpose_w(const float* W, float* Wt, int K, int N, hipStream_t stream) {
  int n = K * N;
  transpose_kernel<<<(n + 255) / 256, 256, 0, stream>>>(W, Wt, K, N);
}

extern "C" void kernel_launch(void* const* d_in, const int* in_sizes, int n_in,
                              void* d_out, int out_size, void* d_ws, size_t ws_size,
                              hipStream_t stream) {
  (void)in_sizes; (void)n_in; (void)out_size; (void)ws_size;
  auto F = [&](int i) { return (const float*)d_in[i]; };
  const float* feat = F(0);
  const float* cond = F(1);
  const float *w1 = F(2), *b1 = F(3), *w2 = F(4), *b2 = F(5), *enh = F(6);
  auto mk = [&](int b) {
    BlockP p;
    p.ln1_g = F(b + 0);  p.ln1_b = F(b + 1);  p.inproj = F(b + 2);
    p.convw = F(b + 3);  p.convb = F(b + 4);  p.xproj  = F(b + 5);
    p.dtw   = F(b + 6);  p.dtb   = F(b + 7);  p.Alog   = F(b + 8);
    p.D     = F(b + 9);  p.ong   = F(b + 10); p.onb    = F(b + 11);
    p.outw  = F(b + 12); p.ln2_g = F(b + 13); p.ln2_b  = F(b + 14);
    p.fc1w  = F(b + 15); p.fc1b  = F(b + 16); p.fc2w   = F(b + 17);
    p.fc2b  = F(b + 18);
    p.inprojT = p.outwT = p.fc1T = p.fc2T = nullptr;
    return p;
  };
  BlockP pm = mk(7), ps = mk(26);

  const size_t L = L_TOK, C = CDIM;
  float* w = (float*)d_ws;
  float* X0  = w;                // feat + cond-injection    (L*C)
  float* XL  = X0 + L * C;       // ps-block output          (L*C)
  float* Y   = XL + L * C;       // pm-block input           (L*C)
  float* CI  = Y + L * C;        // bilerp'd cond            (L*32)
  float* S   = CI + L * 32;      // shared block scratch (sized for pm)
  float* ln1 = S;                // L*64
  float* xz  = ln1 + L * 64;     // L*256
  float* xxc = xz + L * 256;     // L*128
  float* dbl = xxc + L * 128;    // L*144
  float* yb  = dbl + L * 144;    // 4*L*128
  float* ym  = yb + 4 * L * 128; // L*128
  float* ya  = ym + L * 128;     // L*128
  float* x1  = ya + L * 128;     // L*64
  float* h0  = x1 + L * 64;      // L*64
  float* h1  = h0 + L * 64;      // L*256
  float* WT  = h1 + L * 256;     // transposed weights region
  // pm transposed weights
  float* pm_inT  = WT;                 // (256 x 64)
  float* pm_outT = pm_inT  + 64 * 256; // (64 x 128)
  float* pm_f1T  = pm_outT + 128 * 64; // (256 x 64)
  float* pm_f2T  = pm_f1T  + 64 * 256; // (64 x 256)
  // ps transposed weights
  float* ps_inT  = pm_f2T  + 256 * 64; // (128 x 64)
  float* ps_outT = ps_inT  + 64 * 128; // (64 x 64)
  float* ps_f1T  = ps_outT + 64 * 64;  // (64 x 64)
  float* ps_f2T  = ps_f1T  + 64 * 64;  // (64 x 64)

  // one-time device transposes (all GEMMs then use contiguous (NxK) B)
  transpose_w(pm.inproj, pm_inT,  64, 256, stream);
  transpose_w(pm.outw,   pm_outT, 128, 64, stream);
  transpose_w(pm.fc1w,   pm_f1T,  64, 256, stream);
  transpose_w(pm.fc2w,   pm_f2T,  256, 64, stream);
  transpose_w(ps.inproj, ps_inT,  64, 128, stream);
  transpose_w(ps.outw,   ps_outT, 64, 64,  stream);
  transpose_w(ps.fc1w,   ps_f1T,  64, 64,  stream);
  transpose_w(ps.fc2w,   ps_f2T,  64, 64,  stream);
  pm.inprojT = pm_inT;  pm.outwT = pm_outT;  pm.fc1T = pm_f1T;  pm.fc2T = pm_f2T;
  ps.inprojT = ps_inT;  ps.outwT = ps_outT;  ps.fc1T = ps_f1T;  ps.fc2T = ps_f2T;

  // --- cond injection: bilerp(32->64, align_corners) -> 1x1 conv -> dw3x3 ---
  bilerp_kernel<<<(L * 32 + 255) / 256, 256, 0, stream>>>(cond, CI);
  conv1x1_kernel<<<(L * C + 255) / 256, 256, 0, stream>>>(CI, w1, b1, ln1);  // ln1 as tmp
  dw3_add_kernel<<<(L * C + 255) / 256, 256, 0, stream>>>(ln1, w2, b2, feat, X0);

  // --- windowed VSS block (ps): di=64, N=8, mr=1 ---
  run_block(X0, XL, ps, 64, 8, 64, 1, ln1, xz, xxc, dbl, yb, ym, ya, x1, h0, h1, stream);

  // --- y = x_local * enh + x ---
  enh_combine_kernel<<<(L * C + 255) / 256, 256, 0, stream>>>(XL, enh, X0, Y);

  // --- full-image VSS block (pm): di=128, N=16, mr=4 -> d_out ---
  run_block(Y, (float*)d_out, pm, 128, 16, 256, 0, ln1, xz, xxc, dbl, yb, ym, ya, x1, h0, h1, stream);
}